// TruncatedFourierExpansion_4724464025881
// MI455X (gfx1250) — compile-verified
//
#include <hip/hip_runtime.h>
#include <hip/hip_bf16.h>

#define NUM_TERMS   64
#define NUM_INT_PTS 4999

typedef __attribute__((ext_vector_type(2))) float v2f;
typedef __attribute__((ext_vector_type(8))) float v8f;

// ---------------------------------------------------------------------------
// Kernel 1: Fourier coefficients via composite Simpson.
//   block k == 0      -> coef[0]                = A0
//   block k in 1..64  -> coef[k] = An_k, coef[64+k] = Bn_k
// ---------------------------------------------------------------------------
__global__ __launch_bounds__(256)
void fourier_coeff_kernel(const float* __restrict__ Tptr,
                          float* __restrict__ coef)
{
    const int   k      = blockIdx.x;                  // harmonic index (0 => A0)
    const float T      = Tptr[0];
    const float two_pi = 6.28318530717958647692f;
    const float h      = T / (float)(NUM_INT_PTS - 1);
    const float w3     = h / 3.0f;
    const float kw     = two_pi * (float)k / T;

    float accC = 0.0f, accS = 0.0f;
    for (int i = threadIdx.x; i < NUM_INT_PTS; i += 256) {
        float frac = (float)i / (float)(NUM_INT_PTS - 1);
        float xs   = (frac - 0.5f) * T;
        float w    = (i == 0 || i == NUM_INT_PTS - 1) ? 1.0f
                   : ((i & 1) ? 4.0f : 2.0f);
        float fw   = expf(sinf(xs)) * (w * w3);
        float sv, cv;
        sincosf(kw * xs, &sv, &cv);                   // k==0 -> cv=1, sv=0
        accC += cv * fw;
        accS += sv * fw;
    }

    __shared__ float sc[256], ss[256];
    sc[threadIdx.x] = accC;
    ss[threadIdx.x] = accS;
    __syncthreads();
    for (int off = 128; off > 0; off >>= 1) {
        if ((int)threadIdx.x < off) {
            sc[threadIdx.x] += sc[threadIdx.x + off];
            ss[threadIdx.x] += ss[threadIdx.x + off];
        }
        __syncthreads();
    }
    if (threadIdx.x == 0) {
        if (k == 0) {
            coef[0] = sc[0] / T;                      // A0
        } else {
            coef[k]             = (2.0f / T) * sc[0]; // An
            coef[NUM_TERMS + k] = (2.0f / T) * ss[0]; // Bn
        }
    }
}

// ---------------------------------------------------------------------------
// Kernel 2: evaluation. Each wave32 owns TWO tiles of 16 x-values per
// iteration (two independent recurrence chains -> fills VALU delay slots
// and keeps the XDL pipe busy).
//   row = lane & 15  -> x row inside a tile (lane m and m+16 share one x)
//   lanes  0..15 carry odd  harmonics (1,3,...,63)  == f32 A-matrix K=0,1
//   lanes 16..31 carry even harmonics (2,4,...,64)  == f32 A-matrix K=2,3
// Harmonics advance by a rotate-by-2phi recurrence (4 FMA / 2 harmonics).
// Accumulation rides V_WMMA_F32_16X16X4_F32: B column 0 carries the
// coefficient 4-vector (A_h,B_h,A_{h+1},B_{h+1}) via lanes 0 and 16; the
// 32 B-fragments are tile-invariant and pinned in registers up front.
// D column 0 (lanes 0/16, VGPR r = row) is the 16 outputs of a tile.
// ---------------------------------------------------------------------------
__global__ __launch_bounds__(256)
void fourier_eval_kernel(const float* __restrict__ xin,
                         const float* __restrict__ Tptr,
                         const float* __restrict__ coef,
                         float* __restrict__ out,
                         int n)
{
    __shared__ float2 P[NUM_TERMS];   // P[h-1] = (An_h, Bn_h), h = 1..64
    __shared__ float  sA0;

    if (threadIdx.x == 0) sA0 = coef[0];
    if (threadIdx.x < NUM_TERMS) {
        P[threadIdx.x] = make_float2(coef[1 + threadIdx.x],
                                     coef[1 + NUM_TERMS + threadIdx.x]);
    }
    __syncthreads();

    const float T  = Tptr[0];
    const float w0 = 6.28318530717958647692f / T;

    const int lane          = threadIdx.x & 31;
    const int row           = lane & 15;     // x row inside tile
    const int hi            = lane >> 4;     // 0: odd harmonics, 1: even
    const int wavesPerBlock = blockDim.x >> 5;
    const int waveGlobal    = blockIdx.x * wavesPerBlock + (threadIdx.x >> 5);
    const int totalWaves    = gridDim.x * wavesPerBlock;
    const int ntiles        = (n + 15) >> 4;
    const int npairs        = (ntiles + 1) >> 1;

    // Tile-invariant B fragments: built once, live in VGPRs for the kernel.
    v2f bfrag[32];
#pragma unroll
    for (int t = 0; t < 32; ++t) {
        float2 p = P[2 * t + hi];            // lanes<16: h=2t+1, lanes>=16: h=2t+2
        bfrag[t].x = (row == 0) ? p.x : 0.0f;
        bfrag[t].y = (row == 0) ? p.y : 0.0f;
    }
    const float a0 = sA0;

    for (int tp = waveGlobal; tp < npairs; tp += totalWaves) {
        const int tile0 = 2 * tp;
        const int tile1 = 2 * tp + 1;

        // --- load the two x's and build both rotation seeds -----------------
        int   xi0 = (tile0 << 4) + row;
        int   xi1 = (tile1 << 4) + row;
        float x0  = (xi0 < n) ? xin[xi0] : 0.0f;
        float x1  = (xi1 < n) ? xin[xi1] : 0.0f;

        float s1a, c1a, s1b, c1b;
        sincosf(w0 * x0, &s1a, &c1a);
        sincosf(w0 * x1, &s1b, &c1b);

        float c2a = c1a * c1a - s1a * s1a, s2a = 2.0f * s1a * c1a; // rot(2phi)
        float c2b = c1b * c1b - s1b * s1b, s2b = 2.0f * s1b * c1b;

        float ca = hi ? c2a : c1a,  sa = hi ? s2a : s1a;  // start: h=1 / h=2
        float cb = hi ? c2b : c1b,  sb = hi ? s2b : s1b;

        v8f d0 = {};
        v8f d1 = {};

        // --- 32 harmonic-pair steps; two independent chains interleaved -----
#pragma unroll
        for (int t = 0; t < 32; ++t) {
            v2f aa;  aa.x = ca;  aa.y = sa;
            d0 = __builtin_amdgcn_wmma_f32_16x16x4_f32(
                     false, aa, false, bfrag[t], (short)0, d0, false, false);
            float cna = ca * c2a - sa * s2a;
            float sna = sa * c2a + ca * s2a;
            ca = cna;  sa = sna;

            v2f ab;  ab.x = cb;  ab.y = sb;
            d1 = __builtin_amdgcn_wmma_f32_16x16x4_f32(
                     false, ab, false, bfrag[t], (short)0, d1, false, false);
            float cnb = cb * c2b - sb * s2b;
            float snb = sb * c2b + cb * s2b;
            cb = cnb;  sb = snb;
        }

        // --- store: D column 0 lives in lanes row==0 (lane 0 rows 0..7,
        //     lane 16 rows 8..15); vectorized b128 stores on the fast path ---
        if (row == 0) {
            int base0 = (tile0 << 4) + (hi << 3);
            if (base0 + 8 <= n) {
                float4* o = (float4*)(out + base0);
                o[0] = make_float4(a0 + d0[0], a0 + d0[1], a0 + d0[2], a0 + d0[3]);
                o[1] = make_float4(a0 + d0[4], a0 + d0[5], a0 + d0[6], a0 + d0[7]);
            } else {
#pragma unroll
                for (int r = 0; r < 8; ++r) {
                    int oi = base0 + r;
                    if (oi < n) out[oi] = a0 + d0[r];
                }
            }
            int base1 = (tile1 << 4) + (hi << 3);
            if (base1 + 8 <= n) {
                float4* o = (float4*)(out + base1);
                o[0] = make_float4(a0 + d1[0], a0 + d1[1], a0 + d1[2], a0 + d1[3]);
                o[1] = make_float4(a0 + d1[4], a0 + d1[5], a0 + d1[6], a0 + d1[7]);
            } else {
#pragma unroll
                for (int r = 0; r < 8; ++r) {
                    int oi = base1 + r;
                    if (oi < n) out[oi] = a0 + d1[r];
                }
            }
        }
    }
}

// ---------------------------------------------------------------------------
extern "C" void kernel_launch(void* const* d_in, const int* in_sizes, int n_in,
                              void* d_out, int out_size, void* d_ws, size_t ws_size,
                              hipStream_t stream)
{
    const float* xin  = (const float*)d_in[0];   // input_feature [X_DIM]
    const float* Tptr = (const float*)d_in[1];   // scalar T
    float*       out  = (float*)d_out;
    float*       coef = (float*)d_ws;            // 129 floats: A0, An[64], Bn[64]
    const int    n    = in_sizes[0];

    // Phase 1: coefficients (65 blocks: A0 + 64 harmonics)
    fourier_coeff_kernel<<<NUM_TERMS + 1, 256, 0, stream>>>(Tptr, coef);

    // Phase 2: evaluation. 8 waves per block, two 16-element tiles per wave
    // per iteration.
    const int ntiles        = (n + 15) >> 4;
    const int npairs        = (ntiles + 1) >> 1;
    const int wavesPerBlock = 8;
    int blocks = (npairs + wavesPerBlock - 1) / wavesPerBlock;
    if (blocks < 1) blocks = 1;
    fourier_eval_kernel<<<blocks, 256, 0, stream>>>(xin, Tptr, coef, out, n);
}